// FusionWordAttention_73521250173208
// MI455X (gfx1250) — compile-verified
//
#include <hip/hip_runtime.h>
#include <hip/hip_bf16.h>

// Problem constants (from reference): B=16, D=64, H=160, W=320, T=20
#define BB   16
#define DD   64
#define NN   51200   // H*W
#define TT   20
#define NBLK 50      // NN / (8 waves * 8 tiles * 16 cols) = 51200/1024

typedef float v2f __attribute__((ext_vector_type(2)));
typedef float v8f __attribute__((ext_vector_type(8)));

#define NEG_INF (-3.402823466e38f)

// t index covered by accumulator slot s (0..15) for a given lane-half.
__device__ __forceinline__ int slot_t(int s, int half) {
    return ((s >> 3) << 4) + half * 8 + (s & 7);
}

// Stage tokens into LDS (32x64, rows >= 20 zeroed) via the CDNA5 async
// global->LDS path (ASYNCcnt-tracked), then make them visible to all waves.
__device__ __forceinline__ void load_tokens_async(float* tok,
                                                  const float* __restrict__ x2,
                                                  int tid) {
    for (int i = tid; i < 32 * DD; i += 256)          // zero the pad rows
        if (i >= TT * DD) tok[i] = 0.f;
    for (int i = tid; i < TT * DD; i += 256) {        // 1280 = 5*256: EXEC all-ones
        unsigned            lds_off = (unsigned)(size_t)(tok + i);
        unsigned long long  gaddr   = (unsigned long long)(size_t)(x2 + i);
        asm volatile("global_load_async_to_lds_b32 %0, %1, off"
                     :: "v"(lds_off), "v"(gaddr) : "memory");
    }
    asm volatile("s_wait_asynccnt 0x0" ::: "memory");
    __syncthreads();
}

// Fetch this lane's two A-operand pairs (tokens) straight from LDS.
// Inline asm keeps them out of long-lived registers (prevents spills).
__device__ __forceinline__ void lds_load_a(const float* tok, int idx0, int idx1,
                                           v2f& a0, v2f& a1) {
    const unsigned o0 = (unsigned)(size_t)(tok + idx0);
    const unsigned o1 = (unsigned)(size_t)(tok + idx1);
    asm volatile("ds_load_b64 %0, %2\n\t"
                 "ds_load_b64 %1, %3\n\t"
                 "s_wait_dscnt 0x0"
                 : "=&v"(a0), "=&v"(a1)
                 : "v"(o0), "v"(o1));
}

// One 16-column score tile with V_WMMA_F32_16X16X4_F32.
// Phase 1: all 32 B-loads issued as a clause into bv[16] (two bases so every
// offset fits the signed-24-bit immediate). Phase 2: 32-WMMA chain with
// A-operands streamed from LDS. bv doubles as the x1 stash for the caller.
__device__ __forceinline__ void tile_scores(const float* __restrict__ colp0,
                                            const float* __restrict__ colp1,
                                            const float* tok,
                                            int lane, v8f& acc0, v8f& acc1,
                                            v2f* bv)
{
    const int kofs = (lane >> 4) << 1;   // K offset: 0 for lanes 0-15, 2 for 16-31
    const int m    = lane & 15;          // A-matrix row within M-tile
    acc0 = (v8f){0.f,0.f,0.f,0.f,0.f,0.f,0.f,0.f};
    acc1 = acc0;
    __builtin_prefetch(colp0 + 16, 0, 0);              // global_prefetch_b8
#pragma unroll
    for (int kk = 0; kk < 8; ++kk) {                   // d in [0,32): base 0
        const int d0 = kk * 4 + kofs;
        bv[kk].x = colp0[(size_t)d0 * NN];
        bv[kk].y = colp0[(size_t)(d0 + 1) * NN];
    }
#pragma unroll
    for (int kk = 8; kk < 16; ++kk) {                  // d in [32,64): base 1
        const int d0 = kk * 4 + kofs - 32;
        bv[kk].x = colp1[(size_t)d0 * NN];
        bv[kk].y = colp1[(size_t)(d0 + 1) * NN];
    }
#pragma unroll
    for (int kk = 0; kk < 16; ++kk) {
        const int d0 = kk * 4 + kofs;
        v2f a0, a1;
        lds_load_a(tok, m * DD + d0, (16 + m) * DD + d0, a0, a1);
        acc0 = __builtin_amdgcn_wmma_f32_16x16x4_f32(false, a0, false, bv[kk],
                                                     (short)0, acc0, false, false);
        acc1 = __builtin_amdgcn_wmma_f32_16x16x4_f32(false, a1, false, bv[kk],
                                                     (short)0, acc1, false, false);
    }
}

// Combine two online-softmax states (m1,s1) <- (m1,s1) + (m2,s2)
__device__ __forceinline__ void softmax_merge(float& m1, float& s1, float m2, float s2) {
    const float nm = fmaxf(m1, m2);
    s1 = s1 * __expf(m1 - nm) + s2 * __expf(m2 - nm);
    m1 = nm;
}

// ---- Pass 1: scores (WMMA) + online per-(b,t) (max, sumexp) block partials ----
__global__ void __launch_bounds__(256)
k_scores_stats(const float* __restrict__ x1,
               const float* __restrict__ x2,
               float* __restrict__ mpart,   // [B*T][NBLK]
               float* __restrict__ spart)   // [B*T][NBLK]
{
    __shared__ float tok[32 * DD];
    __shared__ float red_m[8][TT];
    __shared__ float red_s[8][TT];
    const int b = blockIdx.y, tid = threadIdx.x, lane = tid & 31, w = tid >> 5;
    load_tokens_async(tok, x2, tid);
    const float* x1b = x1 + (size_t)b * DD * NN;
    float rm[16], rs[16];
#pragma unroll
    for (int s = 0; s < 16; ++s) { rm[s] = NEG_INF; rs[s] = 0.f; }
    const int nw = blockIdx.x * 1024 + w * 128;
    for (int j = 0; j < 8; ++j) {
        const float* colp0 = x1b + (size_t)(nw + j * 16 + (lane & 15));
        const float* colp1 = colp0 + (size_t)32 * NN;
        v8f a0, a1;
        v2f bv[16];
        tile_scores(colp0, colp1, tok, lane, a0, a1, bv);
#pragma unroll
        for (int i = 0; i < 8; ++i) {
            float v = a0[i], nm = fmaxf(rm[i], v);
            rs[i] = rs[i] * __expf(rm[i] - nm) + __expf(v - nm);
            rm[i] = nm;
            v  = a1[i]; nm = fmaxf(rm[8 + i], v);
            rs[8 + i] = rs[8 + i] * __expf(rm[8 + i] - nm) + __expf(v - nm);
            rm[8 + i] = nm;
        }
    }
    const int half = lane >> 4;
#pragma unroll
    for (int s = 0; s < 16; ++s) {
        float m = rm[s], sv = rs[s];
        for (int d = 1; d < 16; d <<= 1) {
            const float om = __shfl_xor(m, d, 32);
            const float os = __shfl_xor(sv, d, 32);
            softmax_merge(m, sv, om, os);
        }
        if ((lane & 15) == 0) {
            const int t = slot_t(s, half);
            if (t < TT) { red_m[w][t] = m; red_s[w][t] = sv; }
        }
    }
    __syncthreads();
    if (tid < TT) {
        float m = red_m[0][tid], sv = red_s[0][tid];
        for (int ww = 1; ww < 8; ++ww) softmax_merge(m, sv, red_m[ww][tid], red_s[ww][tid]);
        mpart[(size_t)(b * TT + tid) * NBLK + blockIdx.x] = m;
        spart[(size_t)(b * TT + tid) * NBLK + blockIdx.x] = sv;
    }
}

// ---- Pass 2 (tiny): merge block partials -> maxv, coef (deterministic) ----
__global__ void k_finalize(const float* __restrict__ mpart,
                           const float* __restrict__ spart,
                           const int* __restrict__ mask,
                           float* __restrict__ maxv,
                           float* __restrict__ coef) {
    const int i = threadIdx.x;            // 0..319 = b*T+t
    const int t = i % TT;
    float M = NEG_INF, Z = 0.f;
    for (int k = 0; k < NBLK; ++k)
        softmax_merge(M, Z, mpart[(size_t)i * NBLK + k], spart[(size_t)i * NBLK + k]);
    float cnt = 0.f;
    for (int j = 0; j < TT; ++j) cnt += (float)mask[j];
    maxv[i] = M;
    coef[i] = (mask[t] != 0) ? (1.f / (Z * cnt)) : 0.f;
}

// ---- Pass 3: weight[b,n] = sum_t coef*exp(s - max); out = x1 * weight ----
__global__ void __launch_bounds__(256)
k_weight_out(const float* __restrict__ x1,
             const float* __restrict__ x2,
             const float* __restrict__ maxv,
             const float* __restrict__ coef,
             float* __restrict__ out)
{
    __shared__ float tok[32 * DD];
    const int b = blockIdx.y, tid = threadIdx.x, lane = tid & 31, w = tid >> 5;
    load_tokens_async(tok, x2, tid);
    const float* x1b = x1 + (size_t)b * DD * NN;
    float* outb      = out + (size_t)b * DD * NN;
    const int half = lane >> 4, col = lane & 15, kofs = half * 2;
    float mx[16], cf[16];
#pragma unroll
    for (int s = 0; s < 16; ++s) {
        const int t = slot_t(s, half);
        if (t < TT) { mx[s] = maxv[b * TT + t]; cf[s] = coef[b * TT + t]; }
        else        { mx[s] = 0.f;              cf[s] = 0.f; }
    }
    const int nw = blockIdx.x * 1024 + w * 128;
    for (int j = 0; j < 8; ++j) {
        const int n = nw + j * 16;
        const float* colp0 = x1b + (size_t)(n + col);
        const float* colp1 = colp0 + (size_t)32 * NN;
        v8f a0, a1;
        v2f bv[16];                        // also serves as the x1 stash
        tile_scores(colp0, colp1, tok, lane, a0, a1, bv);
        float wsum = 0.f;
#pragma unroll
        for (int i = 0; i < 8; ++i) {
            wsum += cf[i]     * __expf(a0[i] - mx[i]);
            wsum += cf[8 + i] * __expf(a1[i] - mx[8 + i]);
        }
        wsum += __shfl_xor(wsum, 16, 32);  // merge the two t-halves (same n)
        float* outp0 = outb + (size_t)(n + col);
        float* outp1 = outp0 + (size_t)32 * NN;
#pragma unroll
        for (int kk = 0; kk < 8; ++kk) {   // exact (d,n) pairs this lane loaded
            const int d0 = kk * 4 + kofs;
            outp0[(size_t)d0 * NN]       = bv[kk].x * wsum;
            outp0[(size_t)(d0 + 1) * NN] = bv[kk].y * wsum;
        }
#pragma unroll
        for (int kk = 8; kk < 16; ++kk) {
            const int d0 = kk * 4 + kofs - 32;
            outp1[(size_t)d0 * NN]       = bv[kk].x * wsum;
            outp1[(size_t)(d0 + 1) * NN] = bv[kk].y * wsum;
        }
    }
}

extern "C" void kernel_launch(void* const* d_in, const int* in_sizes, int n_in,
                              void* d_out, int out_size, void* d_ws, size_t ws_size,
                              hipStream_t stream) {
    const float* x1   = (const float*)d_in[0];   // [B, D, H, W] fp32
    const float* x2   = (const float*)d_in[1];   // [1, T, D]    fp32
    const int*   mask = (const int*)d_in[2];     // [1, T]       int32
    float* out = (float*)d_out;

    // Workspace layout (floats): mpart[320*50] | spart[320*50] | maxv[320] | coef[320]
    float* mpart = (float*)d_ws;
    float* spart = mpart + BB * TT * NBLK;
    float* maxv  = spart + BB * TT * NBLK;
    float* coef  = maxv + BB * TT;

    dim3 grid(NBLK, BB), blk(256);
    k_scores_stats<<<grid, blk, 0, stream>>>(x1, x2, mpart, spart);
    k_finalize    <<<1, BB * TT, 0, stream>>>(mpart, spart, mask, maxv, coef);
    k_weight_out  <<<grid, blk, 0, stream>>>(x1, x2, maxv, coef, out);
}